// PQEmbedding_40492951666901
// MI455X (gfx1250) — compile-verified
//
#include <hip/hip_runtime.h>
#include <stdint.h>

// Problem constants (from reference)
#define NUM_EMB   50000
#define EMB_DIM   1024
#define N_CENT    256
#define BLK       8          // floats per PQ block
#define N_BLK     128        // blocks per embedding row

typedef float        v4f   __attribute__((ext_vector_type(4)));
typedef unsigned int u32x4 __attribute__((ext_vector_type(4)));
typedef int          i32x8 __attribute__((ext_vector_type(8)));
typedef int          i32x4 __attribute__((ext_vector_type(4)));

// One wave (wave32) per token. 8 waves / 256 threads per workgroup.
// Stores: iteration k, lane l writes float4 #(k*32+l) of the 4KB output row
//   -> 32 lanes x 16B contiguous = fully coalesced 512B per store, th:NT.
// Codes: lane l owns codes for blocks l, l+32, l+64, l+96 (4 scattered loads,
//   L2-resident table). The code for float4 #F (block F/2) is pulled from the
//   owning lane with ds_bpermute_b32.
__global__ __launch_bounds__(256) void pq_embed_kernel(
    const float* __restrict__ centroids,     // [256, 8] f32
    const int*   __restrict__ assignments,   // [128 * 50000] codes
    const int*   __restrict__ input_ids,     // [tokens]
    float*       __restrict__ out)           // [tokens, 1024] f32
{
    __shared__ __align__(16) float lds_cent[N_CENT * BLK];   // 8 KB

#if __has_builtin(__builtin_amdgcn_tensor_load_to_lds)
    // Stage the 8KB centroid table into LDS with the Tensor Data Mover.
    // D# per CDNA5 ISA ch.8: 1-D tile of 2048 f32 elements (data_size=4B),
    // tensor_dim0 = tile_dim0 = 2048, tensor_dim1 = tile_dim1 = 1,
    // dim0 stride = 2048, group0.type = 2 ("image"), count = 1.
    // Toolchain is clang-23 (therock): 6-arg builtin
    //   (u32x4 g0, i32x8 g1, i32x4 g2, i32x4 g3, i32x8 pad, i32 cpol).
    if (threadIdx.x == 0) {
        uint64_t ga = (uint64_t)(uintptr_t)centroids;
        uint32_t lo = (uint32_t)(uintptr_t)lds_cent;  // low 32 bits = LDS byte offset
        u32x4 g0 = { 1u,                              // count=1, user mode
                     lo,                              // lds_addr
                     (uint32_t)ga,                    // global_addr[31:0]
                     (uint32_t)(ga >> 32) | (2u << 30) };  // addr[56:32] | type=2
        i32x8 g1 = { (2 << 16),                // data_size = 4B
                     (int)(2048u << 16),       // tensor_dim0[15:0] << 16
                     (1 << 16),                // tensor_dim0[31:16]=0 | tensor_dim1[15:0]=1
                     (int)(2048u << 16),       // tensor_dim1 hi=0 | tile_dim0 = 2048
                     1,                        // tile_dim1 = 1, tile_dim2 = 0
                     2048,                     // tensor_dim0_stride[31:0]
                     0, 0 };                   // stride0 hi, stride1
        i32x4 z4 = { 0, 0, 0, 0 };
        i32x8 z8 = { 0, 0, 0, 0, 0, 0, 0, 0 };
        __builtin_amdgcn_tensor_load_to_lds(g0, g1, z4, z4, z8, 0);
        __builtin_amdgcn_s_wait_tensorcnt(0); // TDM completion is per-issuing-wave
    }
#else
    {   // Fallback: cooperative vectorized copy (512 float4).
        const v4f* src = (const v4f*)centroids;
        v4f*       dst = (v4f*)lds_cent;
        for (int i = threadIdx.x; i < (N_CENT * BLK) / 4; i += 256) dst[i] = src[i];
    }
#endif
    __syncthreads();

    const int lane  = threadIdx.x & 31;
    const int wave  = threadIdx.x >> 5;
    const int token = blockIdx.x * 8 + wave;      // grid sized so this is in-range

    // Token id is wave-uniform: force it scalar.
    const int id = __builtin_amdgcn_readfirstlane(input_ids[token]);

    // Gather this lane's 4 codes (blocks lane, lane+32, lane+64, lane+96).
    // Scattered 4B loads, but the 25.6MB table stays resident in the 192MB L2
    // because the output path below uses non-temporal stores.
    int code[4];
#pragma unroll
    for (int r = 0; r < 4; ++r)
        code[r] = assignments[(lane + 32 * r) * NUM_EMB + id];

    const int half = lane & 1;   // which 16B half of the 32B centroid
    const int s    = lane >> 1;  // sub-index within the 16 blocks of this iter
    v4f* orow = (v4f*)(out + (size_t)token * EMB_DIM);

#pragma unroll
    for (int r = 0; r < 4; ++r) {
#pragma unroll
        for (int j = 0; j < 2; ++j) {
            const int k = 2 * r + j;                       // store iteration 0..7
            // block B = k*16 + (lane>>1); owner lane = (j<<4)|s, register r.
            const int c = __builtin_amdgcn_ds_bpermute((((j << 4) | s) << 2), code[r]);
            v4f v = *(const v4f*)&lds_cent[c * BLK + half * 4];   // ds_load_b128
            __builtin_nontemporal_store(v, orow + (k * 32 + lane)); // b128, th:NT
        }
    }
}

extern "C" void kernel_launch(void* const* d_in, const int* in_sizes, int n_in,
                              void* d_out, int out_size, void* d_ws, size_t ws_size,
                              hipStream_t stream) {
    const float* centroids   = (const float*)d_in[0];
    const int*   assignments = (const int*)d_in[1];
    const int*   input_ids   = (const int*)d_in[2];
    float*       out         = (float*)d_out;

    const int tokens = in_sizes[2];          // 16384 (multiple of 8)
    const int blocks = tokens / 8;           // one wave per token, 8 waves/block
    pq_embed_kernel<<<blocks, 256, 0, stream>>>(centroids, assignments, input_ids, out);
}